// SignedGCNModel_74002286510428
// MI455X (gfx1250) — compile-verified
//
#include <hip/hip_runtime.h>
#include <hip/hip_bf16.h>

// ---------------------------------------------------------------------------
// SignedGCN (2-layer, pos/neg edge sets) for MI455X (gfx1250, wave32).
//
// Roofline: edge scatter dominates (~2.6 GB traffic -> ~110us @ 23.3 TB/s);
// GEMMs are ~1 GFLOP total. Full fp32 precision; dense x@W stages use the
// native fp32 matrix op V_WMMA_F32_16X16X4_F32 with W staged in LDS.
// ---------------------------------------------------------------------------

typedef __attribute__((ext_vector_type(2))) float v2f;
typedef __attribute__((ext_vector_type(8))) float v8f;

// ---------------- scratch zeroing ----------------
__global__ void zero_f32(float* __restrict__ p, long long count) {
  long long t = (long long)blockIdx.x * blockDim.x + threadIdx.x;
  if (t < count) p[t] = 0.0f;
}

// ---------------- degree / norm ----------------
__global__ void degree_accum(const int* __restrict__ dst, int E, float* __restrict__ deg) {
  int t = blockIdx.x * blockDim.x + threadIdx.x;
  if (t < E) atomicAdd(&deg[dst[t]], 1.0f);
}

// dinv = rsqrt(deg_from_edges + 1)  (self loop always makes deg > 0)
__global__ void degree_to_dinv(float* __restrict__ deg, long long count) {
  long long t = (long long)blockIdx.x * blockDim.x + threadIdx.x;
  if (t < count) deg[t] = rsqrtf(deg[t] + 1.0f);
}

// ---------------- fp32 WMMA GEMM: out[n,N] = A[n,K] @ W[K,N] ----------------
// One wave per 16x16 output tile; W staged in LDS; K,N compile-time so the
// k-loop fully unrolls into clauses of global_load_b64 + ds_load + v_wmma.
// A-fragment (16x4 f32, ISA 7.12.2): lanes 0-15 hold M=lane, VGPR{0,1}=K{0,1};
// lanes 16-31 hold M=lane-16, VGPR{0,1}=K{2,3}. B mirrored with N across lanes.
// D (16x16 f32): VGPR v, lanes 0-15 -> M=v, lanes 16-31 -> M=v+8.
template <int K, int N>
__global__ void gemm_wmma_f32(const float* __restrict__ A, const float* __restrict__ W,
                              float* __restrict__ out, int n) {
  __shared__ float Ws[K * N];
  for (int i = threadIdx.x; i < K * N; i += blockDim.x) Ws[i] = W[i];
  __syncthreads();

  int wave = (blockIdx.x * blockDim.x + threadIdx.x) >> 5;
  int lane = threadIdx.x & 31;
  constexpr int colTiles = N >> 4;
  int rowTiles = (n + 15) >> 4;
  if (wave >= rowTiles * colTiles) return;   // wave-uniform: EXEC all-1s inside
  int rowTile = wave / colTiles;
  int colTile = wave - rowTile * colTiles;
  int row0 = rowTile << 4;
  int half = lane >> 4;                      // selects K pair within 16x16x4 step
  int l    = lane & 15;
  // Clamp (not predicate): out row M depends only on A row M, and rows >= n
  // are never stored, so clamping is safe and keeps loads unconditional.
  int arow = min(row0 + l, n - 1);
  int bcol = (colTile << 4) + l;
  const float* __restrict__ aRow = A + (long long)arow * K;

  v8f acc = {};
#pragma unroll
  for (int k0 = 0; k0 < K; k0 += 4) {
    int ka = k0 + (half << 1);
    v2f a = *(const v2f*)(aRow + ka);        // 8B-aligned global_load_b64
    v2f b;
    b.x = Ws[ka * N + bcol];                 // ds_load (W resident in LDS)
    b.y = Ws[(ka + 1) * N + bcol];
    acc = __builtin_amdgcn_wmma_f32_16x16x4_f32(
        /*neg_a=*/false, a, /*neg_b=*/false, b,
        /*c_mod=*/(short)0, acc, /*reuse_a=*/false, /*reuse_b=*/false);
  }
#pragma unroll
  for (int v = 0; v < 8; ++v) {
    int r = row0 + v + (half << 3);
    if (r < n) out[(long long)r * N + bcol] = acc[v];
  }
}

// ---- edge scatter: agg[dst] += h[src] * dinv[src]*dinv[dst], float4/thread --
template <int F>
__global__ void edge_scatter(const float* __restrict__ h, const int* __restrict__ src,
                             const int* __restrict__ dst, const float* __restrict__ dinv,
                             float* __restrict__ agg, int E) {
  constexpr int V = F / 4;                   // float4 chunks per edge (4 or 8)
  long long t = (long long)blockIdx.x * blockDim.x + threadIdx.x;
  if (t >= (long long)E * V) return;
  int e = (int)(t / V);                      // V is a power of 2 -> shifts
  int c = (int)(t - (long long)e * V);
  int s = src[e];
  int d = dst[e];
  float norm = dinv[s] * dinv[d];
  const float4 hv = *(const float4*)(h + (long long)s * F + c * 4);  // 16B aligned
  float* ap = agg + (long long)d * F + c * 4;
  atomicAdd(ap + 0, hv.x * norm);
  atomicAdd(ap + 1, hv.y * norm);
  atomicAdd(ap + 2, hv.z * norm);
  atomicAdd(ap + 3, hv.w * norm);
}

// ---------------- layer-1 finalize: h = relu(pos) - relu(neg) ----------------
__global__ void finalize1(const float* __restrict__ aggp, const float* __restrict__ aggn,
                          const float* __restrict__ hp, const float* __restrict__ hn,
                          const float* __restrict__ dinvp, const float* __restrict__ dinvn,
                          const float* __restrict__ b1p, const float* __restrict__ b1n,
                          float* __restrict__ h, int n) {
  long long t = (long long)blockIdx.x * blockDim.x + threadIdx.x;
  if (t >= (long long)n * 16) return;
  int i = (int)(t >> 4);
  int f = (int)(t & 15);
  float sp = dinvp[i], sn = dinvn[i];
  float vp = aggp[t] + hp[t] * sp * sp + b1p[f];   // self-loop + bias
  float vn = aggn[t] + hn[t] * sn * sn + b1n[f];
  h[t] = fmaxf(vp, 0.0f) - fmaxf(vn, 0.0f);
}

// ---------------- layer-2 finalize + log_softmax (1 wave per node) ----------
__global__ void finalize2_logsoftmax(const float* __restrict__ aggp, const float* __restrict__ aggn,
                                     const float* __restrict__ gp, const float* __restrict__ gn,
                                     const float* __restrict__ dinvp, const float* __restrict__ dinvn,
                                     const float* __restrict__ b2p, const float* __restrict__ b2n,
                                     float* __restrict__ out, int n) {
  int wave = (blockIdx.x * blockDim.x + threadIdx.x) >> 5;
  int lane = threadIdx.x & 31;     // lane == class (C = 32 == wave32)
  if (wave >= n) return;
  long long idx = (long long)wave * 32 + lane;
  float sp = dinvp[wave], sn = dinvn[wave];
  float vp = aggp[idx] + gp[idx] * sp * sp + b2p[lane];
  float vn = aggn[idx] + gn[idx] * sn * sn + b2n[lane];
  float z = fmaxf(vp, 0.0f) - fmaxf(vn, 0.0f);
  float m = z;
#pragma unroll
  for (int off = 16; off; off >>= 1) m = fmaxf(m, __shfl_xor(m, off, 32));
  float s = expf(z - m);
#pragma unroll
  for (int off = 16; off; off >>= 1) s += __shfl_xor(s, off, 32);
  out[idx] = z - m - logf(s);
}

// ---------------------------------------------------------------------------
extern "C" void kernel_launch(void* const* d_in, const int* in_sizes, int n_in,
                              void* d_out, int out_size, void* d_ws, size_t ws_size,
                              hipStream_t stream) {
  const float* x       = (const float*)d_in[0];
  const int*   ei_pos  = (const int*)d_in[1];
  const int*   ei_neg  = (const int*)d_in[2];
  const float* W1p = (const float*)d_in[3];
  const float* b1p = (const float*)d_in[4];
  const float* W1n = (const float*)d_in[5];
  const float* b1n = (const float*)d_in[6];
  const float* W2p = (const float*)d_in[7];
  const float* b2p = (const float*)d_in[8];
  const float* W2n = (const float*)d_in[9];
  const float* b2n = (const float*)d_in[10];
  float* out = (float*)d_out;

  const int n = in_sizes[0] / 128;
  const int E = in_sizes[1] / 2;
  const int* src_pos = ei_pos;      const int* dst_pos = ei_pos + E;
  const int* src_neg = ei_neg;      const int* dst_neg = ei_neg + E;

  // workspace layout (floats), total 210*n ~ 84 MB
  float* ws = (float*)d_ws;
  const long long nn = n;
  float* dinvp = ws;                 // n   (deg accum, then dinv in place)
  float* dinvn = ws + nn;            // n
  float* hp    = ws + 2 * nn;        // 16n
  float* hn    = hp + 16 * nn;       // 16n
  float* agg1p = hn + 16 * nn;       // 16n
  float* agg1n = agg1p + 16 * nn;    // 16n
  float* h     = agg1n + 16 * nn;    // 16n
  float* gp    = h + 16 * nn;        // 32n
  float* gn    = gp + 32 * nn;       // 32n
  float* agg2p = gn + 32 * nn;       // 32n
  float* agg2n = agg2p + 32 * nn;    // 32n

  const int B = 256;
  auto blocks = [](long long work, int b) { return (unsigned)((work + b - 1) / b); };

  // 0) zero atomic targets (deterministic re-run under graph replay)
  zero_f32<<<blocks(2 * nn, B), B, 0, stream>>>(dinvp, 2 * nn);
  zero_f32<<<blocks(32 * nn, B), B, 0, stream>>>(agg1p, 32 * nn);
  zero_f32<<<blocks(64 * nn, B), B, 0, stream>>>(agg2p, 64 * nn);

  // 1) degrees -> dinv
  degree_accum<<<blocks(E, B), B, 0, stream>>>(dst_pos, E, dinvp);
  degree_accum<<<blocks(E, B), B, 0, stream>>>(dst_neg, E, dinvn);
  degree_to_dinv<<<blocks(2 * nn, B), B, 0, stream>>>(dinvp, 2 * nn);

  // 2) layer-1 GEMMs (fp32 WMMA): hp = x@W1p, hn = x@W1n   [n,128]@[128,16]
  {
    long long tiles = (long long)((n + 15) / 16) * 1;
    unsigned g = blocks(tiles * 32, B);  // 8 waves/block, 1 tile/wave
    gemm_wmma_f32<128, 16><<<g, B, 0, stream>>>(x, W1p, hp, n);
    gemm_wmma_f32<128, 16><<<g, B, 0, stream>>>(x, W1n, hn, n);
  }

  // 3) layer-1 edge aggregation (float4 per thread)
  edge_scatter<16><<<blocks((long long)E * 4, B), B, 0, stream>>>(hp, src_pos, dst_pos, dinvp, agg1p, E);
  edge_scatter<16><<<blocks((long long)E * 4, B), B, 0, stream>>>(hn, src_neg, dst_neg, dinvn, agg1n, E);

  // 4) h = relu(pos conv) - relu(neg conv)
  finalize1<<<blocks(nn * 16, B), B, 0, stream>>>(agg1p, agg1n, hp, hn, dinvp, dinvn, b1p, b1n, h, n);

  // 5) layer-2 GEMMs: gp = h@W2p, gn = h@W2n   [n,16]@[16,32]
  {
    long long tiles = (long long)((n + 15) / 16) * 2;
    unsigned g = blocks(tiles * 32, B);
    gemm_wmma_f32<16, 32><<<g, B, 0, stream>>>(h, W2p, gp, n);
    gemm_wmma_f32<16, 32><<<g, B, 0, stream>>>(h, W2n, gn, n);
  }

  // 6) layer-2 edge aggregation
  edge_scatter<32><<<blocks((long long)E * 8, B), B, 0, stream>>>(gp, src_pos, dst_pos, dinvp, agg2p, E);
  edge_scatter<32><<<blocks((long long)E * 8, B), B, 0, stream>>>(gn, src_neg, dst_neg, dinvn, agg2n, E);

  // 7) finalize + log_softmax (wave per node, 32 classes == 32 lanes)
  finalize2_logsoftmax<<<blocks(nn * 32, B), B, 0, stream>>>(agg2p, agg2n, gp, gn, dinvp, dinvn,
                                                             b2p, b2n, out, n);
}